// GPT2Attention_49134425867050
// MI455X (gfx1250) — compile-verified
//
#include <hip/hip_runtime.h>
#include <cmath>

#define NUM_HEADS 16
#define HEAD_DIM 128
#define HDIM 2048
#define QKV_N 6144
#define BATCH 16
#define BS 16
#define T_MAX 2048
#define BLOCKS_PER_SEQ 128

typedef __attribute__((ext_vector_type(2))) float v2f;
typedef __attribute__((ext_vector_type(4))) float v4f;
typedef __attribute__((ext_vector_type(8))) float v8f;

#define WMMA_F32(a, b, c) \
    __builtin_amdgcn_wmma_f32_16x16x4_f32(false, (a), false, (b), (short)0, (c), false, false)

// Load a 16-K chunk (4 WMMA fragments) of A and W.
//   A 16x4 : VGPR v, lanes0-15 -> K=v, lanes16-31 -> K=v+2 (M = lane&15)
//   B 4x16 : VGPR v, lanes0-15 -> K=v, lanes16-31 -> K=v+2 (N = lane&15)
// aRow/wCol are pre-offset by this lane's (m / n, +2*half) position.
template <int N>
__device__ __forceinline__ void load_chunk(const float* __restrict__ aRow,
                                           const float* __restrict__ wCol,
                                           int kk, v2f* av, v2f* bv) {
#pragma unroll
    for (int u = 0; u < 4; ++u) {
        av[u] = *(const v2f*)(aRow + kk + 4 * u);
        const float* wp = wCol + (kk + 4 * u) * N;
        bv[u].x = wp[0];
        bv[u].y = wp[N];
    }
}

// -------------------------------------------------------------------------
// C[16 x N] = A[16 x K] @ W[K x N] + bias
// One block (4 waves) per 16-column tile of C; K split across the 4 waves;
// double-buffered 16-K chunks (4 x V_WMMA_F32_16X16X4_F32 per chunk) so the
// next chunk's 12 loads are in flight while the current chunk's WMMAs run.
// Weights stay RT-cached: W_qkv + W_proj = 67 MB < 192 MB L2.
// -------------------------------------------------------------------------
template <int K, int N>
__global__ __launch_bounds__(128) void gemm16_wmma(const float* __restrict__ A,
                                                   const float* __restrict__ W,
                                                   const float* __restrict__ bias,
                                                   float* __restrict__ C) {
    __shared__ float red[4 * 256];

    const int tid  = threadIdx.x;
    const int wave = __builtin_amdgcn_readfirstlane(tid >> 5);  // scalar -> uniform loop
    const int lane = tid & 31;
    const int half = lane >> 4;  // 0: lanes 0-15, 1: lanes 16-31
    const int l16  = lane & 15;
    const int n0   = blockIdx.x * 16;

    constexpr int KW    = K / 4;    // K range per wave
    constexpr int NSTEP = KW / 16;  // 16-K chunks per wave
    static_assert(NSTEP % 2 == 0, "pipeline needs even step count");
    const int kBeg = wave * KW;

    const float* aRow = A + l16 * K + kBeg + 2 * half;
    const float* wCol = W + (kBeg + 2 * half) * N + n0 + l16;

    v8f acc = {0.f, 0.f, 0.f, 0.f, 0.f, 0.f, 0.f, 0.f};
    v2f a0[4], b0[4], a1[4], b1[4];

    load_chunk<N>(aRow, wCol, 0, a0, b0);
#pragma unroll 1
    for (int s = 0; s < NSTEP; s += 2) {
        load_chunk<N>(aRow, wCol, (s + 1) * 16, a1, b1);
#pragma unroll
        for (int u = 0; u < 4; ++u) acc = WMMA_F32(a0[u], b0[u], acc);
        if (s + 2 < NSTEP) load_chunk<N>(aRow, wCol, (s + 2) * 16, a0, b0);
#pragma unroll
        for (int u = 0; u < 4; ++u) acc = WMMA_F32(a1[u], b1[u], acc);
    }

    // C 16x16: VGPR v, lanes0-15 -> M=v, lanes16-31 -> M=v+8 (N = lane&15)
    *(v8f*)(red + wave * 256 + lane * 8) = acc;
    __syncthreads();

    for (int e = tid; e < 256; e += 128) {
        float s = red[e] + red[256 + e] + red[512 + e] + red[768 + e];
        int ln  = e >> 3;
        int v   = e & 7;
        int row = v + 8 * (ln >> 4);
        int col = n0 + (ln & 15);
        C[row * N + col] = s + bias[col];
    }
}

// -------------------------------------------------------------------------
// Paged attention (decode step). One workgroup of 256 threads per (b, h).
// New token's k/v (position seq_len-1) comes from the QKV projection in LDS
// instead of mutating the read-only caches — identical math to the
// reference's reshape_and_cache + gather. K/V global reads are non-temporal
// (536 MB streamed once; keep L2 for the resident GEMM weights).
// -------------------------------------------------------------------------
__global__ __launch_bounds__(256) void paged_attn(const float* __restrict__ qkv,
                                                  const float* __restrict__ kcache,
                                                  const float* __restrict__ vcache,
                                                  const int* __restrict__ block_table,
                                                  const int* __restrict__ seq_lens,
                                                  float* __restrict__ attn_out) {
    __shared__ __align__(16) float s_scores[T_MAX];
    __shared__ __align__(16) float s_knew[HEAD_DIM];
    __shared__ __align__(16) float s_vnew[HEAD_DIM];
    __shared__ __align__(16) float s_part[8 * HEAD_DIM];
    __shared__ int   s_bt[BLOCKS_PER_SEQ];
    __shared__ float s_red[256];

    const int h   = blockIdx.x;
    const int b   = blockIdx.y;
    const int tid = threadIdx.x;
    const int L   = seq_lens[b];
    const int p   = L - 1;  // position of the freshly projected token

    // ---- stage block table and new-token k/v ----
    if (tid < BLOCKS_PER_SEQ) s_bt[tid] = block_table[b * BLOCKS_PER_SEQ + tid];
    if (tid < HEAD_DIM) {
        s_knew[tid] = qkv[b * QKV_N + HDIM     + h * HEAD_DIM + tid];
        s_vnew[tid] = qkv[b * QKV_N + 2 * HDIM + h * HEAD_DIM + tid];
    }
    for (int t = tid; t < T_MAX; t += 256) s_scores[t] = -INFINITY;
    __syncthreads();

    // ---- phase 1: scores[t] = scale * (q . k_t) ----
    // 8 lanes per token (16 dims each, b128 loads -> 128 B coalesced per group)
    const int lane = tid & 31;
    const int wv   = tid >> 5;
    const int sub  = lane & 7;
    const int tok  = lane >> 3;
    const float scale = 0.08838834764831845f;  // 1/sqrt(128)

    const v4f* qp = (const v4f*)(qkv + b * QKV_N + h * HEAD_DIM + sub * 16);
    v4f q0 = qp[0], q1 = qp[1], q2 = qp[2], q3 = qp[3];

    for (int t = wv * 4 + tok; t < L; t += 32) {
        v4f k0, k1, k2, k3;
        if (t == p) {
            const v4f* kp = (const v4f*)(s_knew + sub * 16);
            k0 = kp[0]; k1 = kp[1]; k2 = kp[2]; k3 = kp[3];
        } else {
            int blk = s_bt[t >> 4];
            const v4f* kp = (const v4f*)(kcache +
                (((blk * BS + (t & 15)) * NUM_HEADS + h) * HEAD_DIM) + sub * 16);
            k0 = __builtin_nontemporal_load(kp + 0);
            k1 = __builtin_nontemporal_load(kp + 1);
            k2 = __builtin_nontemporal_load(kp + 2);
            k3 = __builtin_nontemporal_load(kp + 3);
        }
        float d = q0.x * k0.x + q0.y * k0.y + q0.z * k0.z + q0.w * k0.w
                + q1.x * k1.x + q1.y * k1.y + q1.z * k1.z + q1.w * k1.w
                + q2.x * k2.x + q2.y * k2.y + q2.z * k2.z + q2.w * k2.w
                + q3.x * k3.x + q3.y * k3.y + q3.z * k3.z + q3.w * k3.w;
        d += __shfl_xor(d, 4, 32);
        d += __shfl_xor(d, 2, 32);
        d += __shfl_xor(d, 1, 32);
        if (sub == 0) s_scores[t] = d * scale;
    }
    __syncthreads();

    // ---- phase 2: softmax (masked via -inf fill) ----
    float lmax = -INFINITY;
    for (int t = tid; t < T_MAX; t += 256) lmax = fmaxf(lmax, s_scores[t]);
    s_red[tid] = lmax;
    __syncthreads();
    for (int off = 128; off > 0; off >>= 1) {
        if (tid < off) s_red[tid] = fmaxf(s_red[tid], s_red[tid + off]);
        __syncthreads();
    }
    const float gmax = s_red[0];
    __syncthreads();

    float lsum = 0.f;
    for (int t = tid; t < T_MAX; t += 256) {
        float e = __expf(s_scores[t] - gmax);  // exp(-inf) = 0 masks t >= L
        s_scores[t] = e;
        lsum += e;
    }
    s_red[tid] = lsum;
    __syncthreads();
    for (int off = 128; off > 0; off >>= 1) {
        if (tid < off) s_red[tid] += s_red[tid + off];
        __syncthreads();
    }
    const float inv = 1.0f / s_red[0];
    __syncthreads();

    // ---- phase 3: out[d] = inv * sum_t p[t] * V[t][d] ----
    // 32 lanes x float4 span a full 512 B V row per wave; 8-way t split.
    const int d4 = (tid & 31) * 4;
    const int g  = tid >> 5;  // 0..7
    v4f acc4 = {0.f, 0.f, 0.f, 0.f};
    for (int t = g; t < L; t += 8) {
        float w = s_scores[t];
        v4f v;
        if (t == p) {
            v = *(const v4f*)(s_vnew + d4);
        } else {
            int blk = s_bt[t >> 4];
            v = __builtin_nontemporal_load((const v4f*)(vcache +
                (((blk * BS + (t & 15)) * NUM_HEADS + h) * HEAD_DIM) + d4));
        }
        acc4.x += w * v.x;
        acc4.y += w * v.y;
        acc4.z += w * v.z;
        acc4.w += w * v.w;
    }
    *(v4f*)(s_part + g * HEAD_DIM + d4) = acc4;
    __syncthreads();
    if (tid < HEAD_DIM) {
        float s = 0.f;
#pragma unroll
        for (int gg = 0; gg < 8; ++gg) s += s_part[gg * HEAD_DIM + tid];
        attn_out[(b * NUM_HEADS + h) * HEAD_DIM + tid] = s * inv;
    }
}

extern "C" void kernel_launch(void* const* d_in, const int* in_sizes, int n_in,
                              void* d_out, int out_size, void* d_ws, size_t ws_size,
                              hipStream_t stream) {
    (void)in_sizes; (void)n_in; (void)out_size; (void)ws_size;
    const float* hs          = (const float*)d_in[0];
    const float* kcache      = (const float*)d_in[1];
    const float* vcache      = (const float*)d_in[2];
    const float* W_qkv       = (const float*)d_in[3];
    const float* b_qkv       = (const float*)d_in[4];
    const float* W_proj      = (const float*)d_in[5];
    const float* b_proj      = (const float*)d_in[6];
    // d_in[7] = slot_mapping (derivable from seq_lens + block_table; unused)
    const int*   block_table = (const int*)d_in[8];
    const int*   seq_lens    = (const int*)d_in[9];
    // d_in[10] = max_seq_len (compile-time constant here)
    float* out = (float*)d_out;

    float* qkv  = (float*)d_ws;         // [16, 6144]
    float* attn = qkv + BATCH * QKV_N;  // [16, 2048]

    // 1) fused QKV projection
    gemm16_wmma<HDIM, QKV_N><<<QKV_N / 16, 128, 0, stream>>>(hs, W_qkv, b_qkv, qkv);
    // 2) paged attention (with virtual scatter of the new token)
    paged_attn<<<dim3(NUM_HEADS, BATCH), 256, 0, stream>>>(qkv, kcache, vcache,
                                                           block_table, seq_lens, attn);
    // 3) output projection
    gemm16_wmma<HDIM, HDIM><<<HDIM / 16, 128, 0, stream>>>(attn, W_proj, b_proj, out);
}